// StochasticTwoLayerGCN_79285096284697
// MI455X (gfx1250) — compile-verified
//
#include <hip/hip_runtime.h>

typedef float v2f __attribute__((ext_vector_type(2)));
typedef float v8f __attribute__((ext_vector_type(8)));

#define IN_F  64
#define HID_F 64
#define OUT_F 32

// ---------------------------------------------------------------------------
// Utility: zero a float buffer
// ---------------------------------------------------------------------------
__global__ void gcn_zero_f32(float* __restrict__ p, int n) {
    int i = blockIdx.x * blockDim.x + threadIdx.x;
    if (i < n) p[i] = 0.0f;
}

// ---------------------------------------------------------------------------
// Degree accumulation: one thread per edge, atomic +1.0 on src/dst degree
// ---------------------------------------------------------------------------
__global__ void gcn_degrees(const int* __restrict__ src, const int* __restrict__ dst,
                            float* __restrict__ dsrc, float* __restrict__ ddst, int n_edges) {
    int e = blockIdx.x * blockDim.x + threadIdx.x;
    if (e < n_edges) {
        atomicAdd(&dsrc[src[e]], 1.0f);
        atomicAdd(&ddst[dst[e]], 1.0f);
    }
}

// deg -> max(deg,1)^(-1/2), in place
__global__ void gcn_norm(float* __restrict__ d, int n) {
    int i = blockIdx.x * blockDim.x + threadIdx.x;
    if (i < n) d[i] = rsqrtf(fmaxf(d[i], 1.0f));
}

// ---------------------------------------------------------------------------
// SpMM scatter: agg[dst] += feat[src] * norm_src[src]
// One thread per (edge, 4-feature chunk); float4 gather, 4x f32 atomic scatter.
// F must be a multiple of 4. Total threads = n_edges * (F/4).
// ---------------------------------------------------------------------------
template <int F>
__global__ void gcn_scatter(const float* __restrict__ feat, const float* __restrict__ nsrc,
                            const int* __restrict__ src, const int* __restrict__ dst,
                            float* __restrict__ agg, int n_edges) {
    const int CH = F >> 2;  // float4 chunks per edge
    int tid = blockIdx.x * blockDim.x + threadIdx.x;
    if (tid >= n_edges * CH) return;
    int e = tid / CH;
    int c = (tid - e * CH) << 2;
    int s = src[e];
    int d = dst[e];
    float ns = nsrc[s];
    float4 v = *(const float4*)(feat + (size_t)s * F + c);
    float* p = agg + (size_t)d * F + c;
    atomicAdd(p + 0, v.x * ns);
    atomicAdd(p + 1, v.y * ns);
    atomicAdd(p + 2, v.z * ns);
    atomicAdd(p + 3, v.w * ns);
}

// ---------------------------------------------------------------------------
// WMMA FP32 GEMM: out[M x N] = epilogue( (rowscale ? rowscale⊙A : A)[M x K] @ W[K x N] + bias )
// N = NT*16.  One wave computes a full 16 x N row-block: the A fragment is
// loaded once per K-step and reused across all NT column tiles (B is tiny and
// L2/L0 resident). Uses V_WMMA_F32_16X16X4_F32, accumulating K in steps of 4.
//
// Fragment layouts (ISA 7.12.2, wave32):
//   A 16x4 f32  : lane l<16 -> row=l,   {V0,V1} = A[row][k..k+1]
//                 lane l>=16-> row=l-16,{V0,V1} = A[row][k+2..k+3]
//   B 4x16  f32 : lane l<16 -> col=l,   {V0,V1} = W[k..k+1][col]
//                 lane l>=16-> col=l-16,{V0,V1} = W[k+2..k+3][col]
//   C/D 16x16   : lane half -> M = half*8 + r (r = VGPR 0..7), N = lane&15
// ---------------------------------------------------------------------------
template <int NT>
__global__ void gcn_gemm_wmma(const float* __restrict__ A, const float* __restrict__ W,
                              const float* __restrict__ bias, const float* __restrict__ rowscale,
                              float* __restrict__ out, int M, int K, int fuse_relu) {
    const int N = NT * 16;
    int wave = (blockIdx.x * blockDim.x + threadIdx.x) >> 5;
    int lane = threadIdx.x & 31;
    int mtiles = M >> 4;
    if (wave >= mtiles) return;  // wave-uniform: EXEC stays all-1 for WMMA

    int row0 = wave << 4;
    int half = lane >> 4;   // 0: K pair {k,k+1}, 1: K pair {k+2,k+3}
    int mn   = lane & 15;   // row index (A) / col index (B,D)

    float rs = rowscale ? rowscale[row0 + mn] : 1.0f;
    const float* arow = A + (size_t)(row0 + mn) * K;

    v8f acc[NT];
#pragma unroll
    for (int t = 0; t < NT; ++t) acc[t] = {};

    for (int k = 0; k < K; k += 4) {
        int ka = k + (half << 1);
        v2f a;
        a.x = arow[ka + 0] * rs;
        a.y = arow[ka + 1] * rs;
#pragma unroll
        for (int t = 0; t < NT; ++t) {
            v2f b;
            b.x = W[(size_t)(ka + 0) * N + t * 16 + mn];
            b.y = W[(size_t)(ka + 1) * N + t * 16 + mn];
            acc[t] = __builtin_amdgcn_wmma_f32_16x16x4_f32(
                /*neg_a=*/false, a, /*neg_b=*/false, b,
                /*c_mod=*/(short)0, acc[t], /*reuse_a=*/false, /*reuse_b=*/false);
        }
    }

#pragma unroll
    for (int t = 0; t < NT; ++t) {
        int col = t * 16 + mn;
        float bv = bias ? bias[col] : 0.0f;
#pragma unroll
        for (int r = 0; r < 8; ++r) {
            int m = (half << 3) + r;
            float v = acc[t][r] + bv;
            if (fuse_relu) v = fmaxf(v, 0.0f);
            out[(size_t)(row0 + m) * N + col] = v;
        }
    }
}

// ---------------------------------------------------------------------------
// Final layer-2 epilogue: out = relu(agg32 * norm_dst[row] + b2[col])
// ---------------------------------------------------------------------------
__global__ void gcn_final(const float* __restrict__ agg, const float* __restrict__ ndst,
                          const float* __restrict__ b2, float* __restrict__ out, int n_nodes) {
    int tid = blockIdx.x * blockDim.x + threadIdx.x;
    if (tid >= n_nodes * OUT_F) return;
    int row = tid / OUT_F;
    int col = tid - row * OUT_F;
    out[tid] = fmaxf(agg[tid] * ndst[row] + b2[col], 0.0f);
}

// ---------------------------------------------------------------------------
// Host-side orchestration
// ---------------------------------------------------------------------------
static inline int cdiv(long long a, int b) { return (int)((a + b - 1) / b); }

extern "C" void kernel_launch(void* const* d_in, const int* in_sizes, int n_in,
                              void* d_out, int out_size, void* d_ws, size_t ws_size,
                              hipStream_t stream) {
    const float* x  = (const float*)d_in[0];
    const int*   ei = (const int*)d_in[1];
    const float* W1 = (const float*)d_in[2];
    const float* b1 = (const float*)d_in[3];
    const float* W2 = (const float*)d_in[4];
    const float* b2 = (const float*)d_in[5];
    float* out = (float*)d_out;

    const int n_nodes = in_sizes[0] / IN_F;
    const int n_edges = in_sizes[1] / 2;
    const int* src = ei;              // edge_index row 0
    const int* dst = ei + n_edges;    // edge_index row 1

    // Workspace layout (floats). Region A (agg64) is reused after the layer-1
    // GEMM to hold g (first half) and agg32 (second half).
    float* ws    = (float*)d_ws;
    float* agg64 = ws;                                     // n_nodes*64
    float* g     = ws;                                     // n_nodes*32 (aliases agg64)
    float* agg32 = ws + (size_t)n_nodes * 32;              // n_nodes*32 (aliases agg64)
    float* h     = ws + (size_t)n_nodes * 64;              // n_nodes*64
    float* nsrc  = ws + (size_t)n_nodes * 128;             // n_nodes
    float* ndst  = nsrc + n_nodes;                         // n_nodes

    const int TPB = 256;

    // 1) degree normalizers
    gcn_zero_f32<<<cdiv(2LL * n_nodes, TPB), TPB, 0, stream>>>(nsrc, 2 * n_nodes);
    gcn_degrees<<<cdiv(n_edges, TPB), TPB, 0, stream>>>(src, dst, nsrc, ndst, n_edges);
    gcn_norm<<<cdiv(2LL * n_nodes, TPB), TPB, 0, stream>>>(nsrc, 2 * n_nodes);

    // 2) layer-1 SpMM: agg64 = segment_sum((x * nsrc)[src] -> dst)
    gcn_zero_f32<<<cdiv((long long)n_nodes * IN_F, TPB), TPB, 0, stream>>>(agg64, n_nodes * IN_F);
    gcn_scatter<IN_F><<<cdiv((long long)n_edges * (IN_F / 4), TPB), TPB, 0, stream>>>(
        x, nsrc, src, dst, agg64, n_edges);

    // 3) layer-1 WMMA GEMM with fused norm_dst / bias / relu:
    //    h = relu((ndst ⊙ agg64) @ W1 + b1)
    {
        int waves = n_nodes >> 4;  // one wave per 16-row block
        gcn_gemm_wmma<HID_F / 16><<<cdiv((long long)waves * 32, TPB), TPB, 0, stream>>>(
            agg64, W1, b1, ndst, h, n_nodes, IN_F, /*relu=*/1);
    }

    // 4) layer-2: shrink features BEFORE the SpMM (segment_sum commutes with @W2):
    //    g = h @ W2   (no bias/relu yet)
    {
        int waves = n_nodes >> 4;
        gcn_gemm_wmma<OUT_F / 16><<<cdiv((long long)waves * 32, TPB), TPB, 0, stream>>>(
            h, W2, nullptr, nullptr, g, n_nodes, HID_F, /*relu=*/0);
    }

    // 5) layer-2 SpMM at 32-wide (half the atomic traffic of 64-wide)
    gcn_zero_f32<<<cdiv((long long)n_nodes * OUT_F, TPB), TPB, 0, stream>>>(agg32, n_nodes * OUT_F);
    gcn_scatter<OUT_F><<<cdiv((long long)n_edges * (OUT_F / 4), TPB), TPB, 0, stream>>>(
        g, nsrc, src, dst, agg32, n_edges);

    // 6) out = relu(agg32 * ndst + b2)
    gcn_final<<<cdiv((long long)n_nodes * OUT_F, TPB), TPB, 0, stream>>>(
        agg32, ndst, b2, out, n_nodes);
}